// FastSpectralClustering_25563645346550
// MI455X (gfx1250) — compile-verified
//
#include <hip/hip_runtime.h>
#include <math.h>

typedef __attribute__((ext_vector_type(2))) float v2f;
typedef __attribute__((ext_vector_type(8))) float v8f;

#define Bc 4
#define Nn 8192
#define Kc 16

// ---------------------------------------------------------------------------
// Kernel 0: zero the small accumulator region of the workspace (StS + scalars)
// ---------------------------------------------------------------------------
__global__ void zero_acc_k(float* __restrict__ p, int n) {
    int i = blockIdx.x * blockDim.x + threadIdx.x;
    if (i < n) p[i] = 0.0f;
}

// ---------------------------------------------------------------------------
// Kernel 1: row softmax over K=16, writes S into d_out (first B*N*K floats)
// ---------------------------------------------------------------------------
__global__ __launch_bounds__(256) void softmax16_k(const float* __restrict__ logits,
                                                   float* __restrict__ S) {
    const int i = blockIdx.x * 256 + threadIdx.x;      // row id, 0..B*N-1
    const float* in = logits + (size_t)i * Kc;
    float v[Kc];
#pragma unroll
    for (int k = 0; k < Kc; ++k) v[k] = in[k];
    float mx = v[0];
#pragma unroll
    for (int k = 1; k < Kc; ++k) mx = fmaxf(mx, v[k]);
    float sum = 0.0f;
#pragma unroll
    for (int k = 0; k < Kc; ++k) { v[k] = __expf(v[k] - mx); sum += v[k]; }
    const float inv = 1.0f / sum;
    float* out = S + (size_t)i * Kc;
#pragma unroll
    for (int k = 0; k < Kc; ++k) out[k] = v[k] * inv;
}

// ---------------------------------------------------------------------------
// Kernel 2: Mt[b, n, k] = (S^T W)[k, n]  via V_WMMA_F32_16X16X4_F32
//   A (16x4, M=cluster, K=row):  lane m in 0..15 -> VGPR0=K0,VGPR1=K1;
//                                lanes 16..31    -> K2,K3
//   B (4x16,  K=row,  N=col):    same row indexing, N = lane&15
//   C/D (16x16): vgpr r, lanes 0-15 -> M=r, lanes 16-31 -> M=r+8
// ---------------------------------------------------------------------------
__device__ __forceinline__ v8f wmma_step(const float* __restrict__ Sb,
                                         const float* __restrict__ Wb,
                                         int n, int lh, int lm, int col0, v8f c) {
    const size_t r = (size_t)(n + 2 * lh);
    v2f a, bm;
    a.x  = Sb[r * Kc + lm];
    a.y  = Sb[(r + 1) * Kc + lm];
    // W is streamed exactly once -> non-temporal to keep S resident in L2
    bm.x = __builtin_nontemporal_load(Wb + r * Nn + col0 + lm);
    bm.y = __builtin_nontemporal_load(Wb + (r + 1) * Nn + col0 + lm);
    return __builtin_amdgcn_wmma_f32_16x16x4_f32(false, a, false, bm,
                                                 (short)0, c, false, false);
}

__global__ __launch_bounds__(256) void stw_gemm_k(const float* __restrict__ W,
                                                  const float* __restrict__ S,
                                                  float* __restrict__ Mt) {
    const int b    = blockIdx.y;
    const int col0 = blockIdx.x << 4;          // 16-column stripe of W
    const int lane = threadIdx.x & 31;
    const int wave = threadIdx.x >> 5;         // 8 waves split the reduction
    const int lh   = lane >> 4;
    const int lm   = lane & 15;

    const float* Wb = W + (size_t)b * Nn * Nn;
    const float* Sb = S + (size_t)b * Nn * Kc;

    // 4 independent accumulator chains to hide WMMA latency
    v8f c0 = {}, c1 = {}, c2 = {}, c3 = {};
    const int n0 = wave * (Nn / 8);
    for (int n = n0; n < n0 + Nn / 8; n += 16) {
        c0 = wmma_step(Sb, Wb, n,      lh, lm, col0, c0);
        c1 = wmma_step(Sb, Wb, n + 4,  lh, lm, col0, c1);
        c2 = wmma_step(Sb, Wb, n + 8,  lh, lm, col0, c2);
        c3 = wmma_step(Sb, Wb, n + 12, lh, lm, col0, c3);
    }
    v8f c = (c0 + c1) + (c2 + c3);

    // cross-wave reduction of the 8 partial 16x16 fp32 tiles
    __shared__ float red[8][8][32];
#pragma unroll
    for (int r = 0; r < 8; ++r) red[wave][r][lane] = c[r];
    __syncthreads();

    const int r = wave;                         // this thread handles (r, lane)
    float v = 0.0f;
#pragma unroll
    for (int w = 0; w < 8; ++w) v += red[w][r][lane];

    const int cl  = r + (lh << 3);              // cluster index 0..15
    const int col = col0 + lm;                  // column in [0, N)
    Mt[((size_t)b * Nn + col) * Kc + cl] = v;   // store transposed: Mt[b][n][k]
}

// ---------------------------------------------------------------------------
// Kernel 3: per-row reductions.
//   trace(cut) = sum_n <Mt[n,:], S[n,:]>,  cut.sum() = sum_n sum_k Mt[n,k]*sum_j S[n,j]
//   StS[b]     = sum_n S[n,:] outer S[n,:]  (lane-staggered LDS atomics)
// ---------------------------------------------------------------------------
__global__ __launch_bounds__(256) void epilogue_k(const float* __restrict__ S,
                                                  const float* __restrict__ Mt,
                                                  float* __restrict__ StS,
                                                  float* __restrict__ scal) {
    __shared__ float sts[Kc * Kc];
    __shared__ float acc[2];
    const int t = threadIdx.x;
    if (t < 2) acc[t] = 0.0f;
    sts[t] = 0.0f;
    __syncthreads();

    const int b = blockIdx.y;
    const int n = blockIdx.x * 256 + t;         // grid.x = Nn/256 (batch-aligned)
    const size_t row = ((size_t)b * Nn + n) * Kc;

    float s[Kc], m[Kc];
#pragma unroll
    for (int k = 0; k < Kc; ++k) { s[k] = S[row + k]; m[k] = Mt[row + k]; }

    float tr = 0.0f, msum = 0.0f, ssum = 0.0f;
#pragma unroll
    for (int k = 0; k < Kc; ++k) { tr += s[k] * m[k]; msum += m[k]; ssum += s[k]; }
    atomicAdd(&acc[0], tr);
    atomicAdd(&acc[1], msum * ssum);            // == row's contribution to cut.sum()

    const int lane = t & 31;
#pragma unroll
    for (int j = 0; j < Kc; ++j) {
        const float sj = s[j];
#pragma unroll
        for (int d = 0; d < Kc; ++d) {
            const int k = (d + lane) & 15;      // stagger by lane: conflict-free
            atomicAdd(&sts[j * Kc + k], sj * s[k]);
        }
    }
    __syncthreads();

    atomicAdd(&StS[b * Kc * Kc + t], sts[t]);
    if (t == 0) { atomicAdd(&scal[0], acc[0]); atomicAdd(&scal[1], acc[1]); }
}

// ---------------------------------------------------------------------------
// Kernel 4: scalar loss
// ---------------------------------------------------------------------------
__global__ void finalize_k(const float* __restrict__ StS,
                           const float* __restrict__ scal,
                           float* __restrict__ loss) {
    if (threadIdx.x == 0 && blockIdx.x == 0) {
        const float cut_loss = scal[0] / scal[1];
        float ss = 0.0f;
        for (int b = 0; b < Bc; ++b)
            for (int j = 0; j < Kc; ++j)
                for (int k = 0; k < Kc; ++k) {
                    float v = StS[b * Kc * Kc + j * Kc + k] - ((j == k) ? 1.0f : 0.0f);
                    ss += v * v;
                }
        loss[0] = -cut_loss + sqrtf(ss) / (float)Bc;
    }
}

// ---------------------------------------------------------------------------
extern "C" void kernel_launch(void* const* d_in, const int* in_sizes, int n_in,
                              void* d_out, int out_size, void* d_ws, size_t ws_size,
                              hipStream_t stream) {
    (void)in_sizes; (void)n_in; (void)out_size; (void)ws_size;
    const float* W      = (const float*)d_in[1];   // [B, N, N]
    const float* logits = (const float*)d_in[2];   // [B, N, K]  (features d_in[0] unused)

    float* out  = (float*)d_out;
    float* S    = out;                              // [B, N, K] -> output 0
    float* loss = out + (size_t)Bc * Nn * Kc;       // scalar    -> output 1

    float* Mt   = (float*)d_ws;                     // [B, N, K] transposed S^T W
    float* StS  = Mt + (size_t)Bc * Nn * Kc;        // [B, K, K]
    float* scal = StS + Bc * Kc * Kc;               // {trace_sum, cut_total}

    zero_acc_k<<<dim3(5), dim3(256), 0, stream>>>(StS, Bc * Kc * Kc + 2);
    softmax16_k<<<dim3((Bc * Nn) / 256), dim3(256), 0, stream>>>(logits, S);
    stw_gemm_k<<<dim3(Nn / 16, Bc), dim3(256), 0, stream>>>(W, S, Mt);
    epilogue_k<<<dim3(Nn / 256, Bc), dim3(256), 0, stream>>>(S, Mt, StS, scal);
    finalize_k<<<dim3(1), dim3(64), 0, stream>>>(StS, scal, loss);
}